// NodePropagator_75110388073048
// MI455X (gfx1250) — compile-verified
//
#include <hip/hip_runtime.h>
#include <cstdint>

// Gather+concat: out[e, 0:128]   = node_states[edge_src[e]]
//                out[e, 128:256] = node_states[edge_tgt[e]]
//
// Pure data movement (0 FLOPs). MI455X strategy:
//  - one wave32 per edge; each lane moves 16B (B128) => one row = one
//    fully-coalesced 512B wave transaction per load/store.
//  - node table (5.12 MB) stays resident in the 192 MB L2 (~64x reuse); the
//    327 MB output stream is written with TH=NT so it cannot evict the table.
//  - unconditional global_prefetch_b8 of the int64 index arrays a few
//    thousand edges ahead (per-lane requests to the same line coalesce;
//    no exec-mask branch needed).

typedef float v4f __attribute__((ext_vector_type(4)));

#ifndef D_FEAT
#define D_FEAT 128
#endif

constexpr int kEdgesPerBlock = 8;    // 8 waves of 32 => 256 threads/block
constexpr int kPrefetchAhead = 4096; // edges of index-array lookahead

__global__ __launch_bounds__(256) void node_propagator_gather_kernel(
    const float* __restrict__ node_states,   // [N, 128]
    const int64_t* __restrict__ edge_src,    // [E]
    const int64_t* __restrict__ edge_tgt,    // [E]
    float* __restrict__ out,                 // [E, 256]
    int n_edges)
{
    const int wave = threadIdx.x >> 5;   // 0..7  : which edge within the block
    const int lane = threadIdx.x & 31;   // 0..31 : which 16B chunk of the row
    const int e = blockIdx.x * kEdgesPerBlock + wave;
    if (e >= n_edges) return;

    // Warm GL2 for the index arrays ahead of use (gfx1250 global_prefetch_b8;
    // speculative => silently dropped past the end of the buffer). All lanes
    // hit the same cacheline; requests coalesce, so no lane guard (keeps the
    // straight-line path branch-free).
    __builtin_prefetch(edge_src + e + kPrefetchAhead, 0, 1);
    __builtin_prefetch(edge_tgt + e + kPrefetchAhead, 0, 1);

    // Wave-uniform index loads (same address across all 32 lanes -> single
    // L2 transaction).
    const int64_t s = edge_src[e];
    const int64_t t = edge_tgt[e];

    // Each lane owns 16 bytes of the 512-byte row.
    const v4f* srow = reinterpret_cast<const v4f*>(
                          node_states + (size_t)s * D_FEAT) + lane;
    const v4f* trow = reinterpret_cast<const v4f*>(
                          node_states + (size_t)t * D_FEAT) + lane;

    // Regular-temporal loads: keep the 5 MB node table hot in L2.
    v4f a = *srow;   // global_load_b128
    v4f b = *trow;   // global_load_b128

    // Nontemporal stores: the 327 MB output is write-once streamed; TH=NT
    // prevents it from evicting the node table from L2.
    v4f* o = reinterpret_cast<v4f*>(out + (size_t)e * (2 * D_FEAT)) + lane;
    __builtin_nontemporal_store(a, o);        // src half: floats [0,128)
    __builtin_nontemporal_store(b, o + 32);   // tgt half: floats [128,256)
}

extern "C" void kernel_launch(void* const* d_in, const int* in_sizes, int n_in,
                              void* d_out, int out_size, void* d_ws, size_t ws_size,
                              hipStream_t stream) {
    const float*   node_states = static_cast<const float*>(d_in[0]);
    const int64_t* edge_src    = static_cast<const int64_t*>(d_in[1]);
    const int64_t* edge_tgt    = static_cast<const int64_t*>(d_in[2]);
    float*         out         = static_cast<float*>(d_out);

    const int n_edges = in_sizes[1];  // E = 320000

    const int blocks = (n_edges + kEdgesPerBlock - 1) / kEdgesPerBlock;
    node_propagator_gather_kernel<<<blocks, 256, 0, stream>>>(
        node_states, edge_src, edge_tgt, out, n_edges);
}